// LFQ_48919677501491
// MI455X (gfx1250) — compile-verified
//
#include <hip/hip_runtime.h>
#include <cstdint>

typedef float v2f __attribute__((ext_vector_type(2)));
typedef float v8f __attribute__((ext_vector_type(8)));

#define B_    16
#define D_    512
#define T_    4096
#define NPOS  (B_ * T_)            // 65536 positions
#define CBSZ  1024
#define CBDIM 10

// ---------------------------------------------------------------------------
// Kernel 1: weight normalization + accumulator zeroing
//   w_in_pad  [16][512]  (rows >= 10 zero)
//   w_out_pad [512][16]  (cols >= 10 zero)
// ---------------------------------------------------------------------------
__global__ __launch_bounds__(256) void lfq_prep(
    const float* __restrict__ in_v, const float* __restrict__ in_g,
    const float* __restrict__ out_v, const float* __restrict__ out_g,
    float* __restrict__ ws)
{
    float* w_in  = ws;                     // 8192 floats
    float* w_out = ws + 8192;              // 8192 floats
    float* accs  = ws + 16384 + 524288;    // 1024 avg-prob bins + 2 scalars
    int tid = threadIdx.x, lane = tid & 31, wv = tid >> 5;

    // one wave per row of w_in_pad (rows wv and wv+8)
    for (int r = wv; r < 16; r += 8) {
        if (r < CBDIM) {
            float s = 0.f;
            for (int i = lane; i < D_; i += 32) { float v = in_v[r * D_ + i]; s += v * v; }
            for (int m = 16; m >= 1; m >>= 1) s += __shfl_xor(s, m, 32);
            float sc = in_g[r] * rsqrtf(s);
            for (int i = lane; i < D_; i += 32) w_in[r * D_ + i] = in_v[r * D_ + i] * sc;
        } else {
            for (int i = lane; i < D_; i += 32) w_in[r * D_ + i] = 0.f;
        }
    }
    // w_out_pad rows (norm over 10 elements)
    for (int o = tid; o < D_; o += 256) {
        float s = 0.f;
        for (int d = 0; d < CBDIM; ++d) { float v = out_v[o * CBDIM + d]; s += v * v; }
        float sc = out_g[o] * rsqrtf(s);
        for (int d = 0; d < 16; ++d)
            w_out[o * 16 + d] = (d < CBDIM) ? out_v[o * CBDIM + d] * sc : 0.f;
    }
    for (int i = tid; i < 1026; i += 256) accs[i] = 0.f;
}

// ---------------------------------------------------------------------------
// Kernel 2: LUT build via WMMA:  lut[j][o] = sum_d w_out[o][d]*(+/-1) + out_b[o]
//   M = 512 (o), N = 1024 (j), K = 16 (padded 10). 2048 waves, one tile each.
// ---------------------------------------------------------------------------
__global__ __launch_bounds__(256) void lfq_lut(
    const float* __restrict__ out_b, const float* __restrict__ ws,
    float* __restrict__ lut)
{
    const float* w_out = ws + 8192;
    int lane = threadIdx.x & 31;
    int wave = blockIdx.x * 8 + (threadIdx.x >> 5);
    int mt = wave & 31;            // o-tile (32 tiles of 16)
    int nt = wave >> 5;            // j-tile (64 tiles of 16)
    int n = lane & 15, h = lane >> 4;
    int j = nt * 16 + n;

    v8f acc = {0.f, 0.f, 0.f, 0.f, 0.f, 0.f, 0.f, 0.f};
#pragma unroll
    for (int k0 = 0; k0 < 16; k0 += 4) {
        int ka = k0 + h * 2;
        v2f av, bv;
        av.x = w_out[(mt * 16 + n) * 16 + ka];
        av.y = w_out[(mt * 16 + n) * 16 + ka + 1];
        bv.x = (ka     < CBDIM) ? (((j >> (9 - ka    )) & 1) ? 1.f : -1.f) : 0.f;
        bv.y = (ka + 1 < CBDIM) ? (((j >> (9 - ka - 1)) & 1) ? 1.f : -1.f) : 0.f;
        acc = __builtin_amdgcn_wmma_f32_16x16x4_f32(false, av, false, bv,
                                                    (short)0, acc, false, false);
    }
#pragma unroll
    for (int r = 0; r < 8; ++r) {
        int o = mt * 16 + r + h * 8;
        lut[(size_t)j * D_ + o] = acc[r] + out_b[o];
    }
}

// ---------------------------------------------------------------------------
// Kernel 3: fused main pass.  512 WGs x 8 waves; wave = one 16-wide t tile.
//   GEMM xp = w_in @ x via v_wmma_f32_16x16x4_f32 (K=512), then per-position
//   sign/index/commit, factorized Bernoulli entropy, and 1024-bin avg-prob
//   histogram (LDS ds_add_f32 atomics -> global atomics).
// ---------------------------------------------------------------------------
__global__ __launch_bounds__(256) void lfq_main(
    const float* __restrict__ x, const float* __restrict__ in_b,
    const float* __restrict__ ws_w, float* __restrict__ avg_acc,
    float* __restrict__ scal, int* __restrict__ idx_ws,
    float* __restrict__ idx_out)
{
    __shared__ float s_w[16 * 512];
    __shared__ float s_stage[8 * 256];
    __shared__ float s_acc[1024];
    __shared__ float s_scal[2];

    int tid = threadIdx.x;
    for (int i = tid; i < 8192; i += 256) s_w[i] = ws_w[i];
    for (int i = tid; i < 1024; i += 256) s_acc[i] = 0.f;
    if (tid < 2) s_scal[tid] = 0.f;
    __syncthreads();

    int wave = tid >> 5, lane = tid & 31;
    int n = lane & 15, h = lane >> 4;
    int wg = blockIdx.x;                    // 512 workgroups
    int b = wg >> 5;                        // 32 WGs per batch element
    int tbase = (wg & 31) * 128 + wave * 16;
    const float* xb = x + ((size_t)b * D_ * T_) + tbase + n;

    v8f acc0 = {0.f, 0.f, 0.f, 0.f, 0.f, 0.f, 0.f, 0.f};
    v8f acc1 = {0.f, 0.f, 0.f, 0.f, 0.f, 0.f, 0.f, 0.f};
#pragma unroll 4
    for (int k0 = 0; k0 < D_; k0 += 8) {
        {
            int ka = k0 + h * 2;
            v2f bv, av;
            bv.x = xb[(size_t)ka * T_];
            bv.y = xb[(size_t)(ka + 1) * T_];
            av.x = s_w[n * D_ + ka];
            av.y = s_w[n * D_ + ka + 1];
            acc0 = __builtin_amdgcn_wmma_f32_16x16x4_f32(false, av, false, bv,
                                                         (short)0, acc0, false, false);
        }
        {
            int ka = k0 + 4 + h * 2;
            v2f bv, av;
            bv.x = xb[(size_t)ka * T_];
            bv.y = xb[(size_t)(ka + 1) * T_];
            av.x = s_w[n * D_ + ka];
            av.y = s_w[n * D_ + ka + 1];
            acc1 = __builtin_amdgcn_wmma_f32_16x16x4_f32(false, av, false, bv,
                                                         (short)0, acc1, false, false);
        }
    }

    // stage C tile to LDS (transpose channels onto per-column lanes)
    float* st = &s_stage[wave * 256];
#pragma unroll
    for (int r = 0; r < 8; ++r) {
        int m = r + h * 8;
        float v = acc0[r] + acc1[r];
        if (m < CBDIM) v += in_b[m];
        st[m * 16 + n] = v;
    }
    asm volatile("s_wait_dscnt 0" ::: "memory");

    // per-column work: column n (both half-waves read it, split the stats work)
    float ov[10], p[10];
#pragma unroll
    for (int d = 0; d < CBDIM; ++d) ov[d] = st[d * 16 + n];

    int idx = 0;
#pragma unroll
    for (int d = 0; d < CBDIM; ++d) idx |= (ov[d] > 0.f ? 1 : 0) << (9 - d);
    int tpos = b * T_ + tbase + n;
    if (h == 0) { idx_ws[tpos] = idx; idx_out[tpos] = (float)idx; }

#pragma unroll
    for (int d = 0; d < CBDIM; ++d) p[d] = 1.f / (1.f + __expf(-400.f * ov[d]));

    // per-sample entropy (factorized) + commit loss, d-range split across halves
    float ent = 0.f, com = 0.f;
#pragma unroll
    for (int dd = 0; dd < 5; ++dd) {
        int d = h * 5 + dd;
        float pd = p[d];
        ent -= pd * __logf(fmaxf(pd, 1e-5f)) + (1.f - pd) * __logf(fmaxf(1.f - pd, 1e-5f));
        float q = ov[d] > 0.f ? 1.f : -1.f;
        float df = ov[d] - q;
        com += df * df;
    }
    for (int m = 16; m >= 1; m >>= 1) {
        ent += __shfl_xor(ent, m, 32);
        com += __shfl_xor(com, m, 32);
    }
    if (lane == 0) { atomicAdd(&s_scal[0], ent); atomicAdd(&s_scal[1], com); }

    // avg-prob histogram: prob_j = prefix(hi bits d0..d4) * suffix(lo bits d5..d9)
    float suf[32];
    suf[0] = 1.f;
    int cnt = 1;
#pragma unroll
    for (int d = 5; d < 10; ++d) {
#pragma unroll
        for (int i2 = cnt - 1; i2 >= 0; --i2) {
            float base2 = suf[i2];
            suf[2 * i2 + 1] = base2 * p[d];
            suf[2 * i2]     = base2 * (1.f - p[d]);
        }
        cnt <<= 1;
    }
    for (int g = 0; g < 16; ++g) {
        int hi = h * 16 + g;                 // each half-wave covers 16 prefixes
        float pref = 1.f;
#pragma unroll
        for (int d = 0; d < 5; ++d)
            pref *= ((hi >> (4 - d)) & 1) ? p[d] : (1.f - p[d]);
        float* ap = &s_acc[hi * 32];
#pragma unroll
        for (int lo = 0; lo < 32; ++lo)
            atomicAdd(&ap[lo], pref * suf[lo]);
    }

    __syncthreads();
    for (int i = tid; i < 1024; i += 256) atomicAdd(&avg_acc[i], s_acc[i]);
    if (tid == 0) { atomicAdd(&scal[0], s_scal[0]); atomicAdd(&scal[1], s_scal[1]); }
}

// ---------------------------------------------------------------------------
// Kernel 4: output stream — gather lut columns (L2-resident) and write out.
// ---------------------------------------------------------------------------
__global__ __launch_bounds__(256) void lfq_out(
    const float* __restrict__ lut, const int* __restrict__ idx,
    float* __restrict__ out)
{
    size_t total  = (size_t)B_ * D_ * T_;
    size_t stride = (size_t)gridDim.x * blockDim.x;
    for (size_t e = (size_t)blockIdx.x * blockDim.x + threadIdx.x; e < total; e += stride) {
        int t  = (int)(e & (T_ - 1));
        int o  = (int)((e >> 12) & (D_ - 1));
        int bb = (int)(e >> 21);
        int id = idx[bb * T_ + t];
        out[e] = lut[(size_t)id * D_ + o];
    }
}

// ---------------------------------------------------------------------------
// Kernel 5: finalize aux loss.
// ---------------------------------------------------------------------------
__global__ __launch_bounds__(256) void lfq_fin(
    const float* __restrict__ avg_acc, const float* __restrict__ scal,
    float* __restrict__ aux)
{
    __shared__ float red[256];
    int tid = threadIdx.x;
    float ce = 0.f;
    for (int i = tid; i < CBSZ; i += 256) {
        float ap = avg_acc[i] * (1.f / (float)NPOS);
        ce -= ap * __logf(fmaxf(ap, 1e-5f));
    }
    red[tid] = ce;
    __syncthreads();
    for (int s = 128; s > 0; s >>= 1) {
        if (tid < s) red[tid] += red[tid + s];
        __syncthreads();
    }
    if (tid == 0) {
        float pse = scal[0] / (float)NPOS;
        float com = scal[1] / (float)(NPOS * CBDIM);
        aux[0] = pse - red[0] + com;   // ENTROPY_W=1, GAMMA=1, COMMIT_W=1
    }
}

// ---------------------------------------------------------------------------
extern "C" void kernel_launch(void* const* d_in, const int* in_sizes, int n_in,
                              void* d_out, int out_size, void* d_ws, size_t ws_size,
                              hipStream_t stream)
{
    (void)in_sizes; (void)n_in; (void)out_size; (void)ws_size;
    const float* x     = (const float*)d_in[0];
    const float* in_v  = (const float*)d_in[1];
    const float* in_g  = (const float*)d_in[2];
    const float* in_b  = (const float*)d_in[3];
    const float* out_v = (const float*)d_in[4];
    const float* out_g = (const float*)d_in[5];
    const float* out_b = (const float*)d_in[6];

    float* out = (float*)d_out;
    float* ws  = (float*)d_ws;
    // workspace layout (floats): w_in_pad[8192] | w_out_pad[8192] | lut[524288]
    //                            | avg_acc[1024] | scal[2] | idx_ws[65536 ints]
    float* w_in    = ws;
    float* lut     = ws + 16384;
    float* avg_acc = ws + 16384 + 524288;
    float* scal    = avg_acc + 1024;
    int*   idx_ws  = (int*)(scal + 2);

    float* idx_out = out + (size_t)B_ * D_ * T_;   // indices (as float), 65536
    float* aux     = idx_out + NPOS;               // scalar aux loss

    lfq_prep<<<1,    256, 0, stream>>>(in_v, in_g, out_v, out_g, ws);
    lfq_lut <<<256,  256, 0, stream>>>(out_b, ws, lut);
    lfq_main<<<512,  256, 0, stream>>>(x, in_b, w_in, avg_acc, scal, idx_ws, idx_out);
    lfq_out <<<8192, 256, 0, stream>>>(lut, idx_ws, out);
    lfq_fin <<<1,    256, 0, stream>>>(avg_acc, scal, aux);
}